// GIN_37898791420018
// MI455X (gfx1250) — compile-verified
//
#include <hip/hip_runtime.h>
#include <hip/hip_bf16.h>

typedef __attribute__((ext_vector_type(2))) float v2f;
typedef __attribute__((ext_vector_type(4))) float v4f;
typedef __attribute__((ext_vector_type(8))) float v8f;

#define GIN_N_NODES 50000
#define GIN_N_EDGES 800000
#define GIN_IN_DIM  128
#define GIN_HID     256
#define GIN_OUT_DIM 64
#define GIN_N_CONV  4
#define GIN_BN_EPS  1e-5f

// ---------------- CDNA5 async global->LDS helpers (ASYNCcnt-tracked DMA) ----------------
__device__ __forceinline__ void async_lds_b128(unsigned lds_byte_off, const float* gptr) {
    // GLOBAL_LOAD_ASYNC_TO_LDS_B128: per-lane 16B memory -> LDS, tracked with ASYNCcnt
    asm volatile("global_load_async_to_lds_b128 %0, %1, off"
                 :: "v"(lds_byte_off), "v"(gptr) : "memory");
}
__device__ __forceinline__ void wait_async_2() { asm volatile("s_wait_asynccnt 0x2" ::: "memory"); }
__device__ __forceinline__ void wait_async_0() { asm volatile("s_wait_asynccnt 0x0" ::: "memory"); }

// ============================ small utility kernels ============================

__global__ void k_zero_i32(int* __restrict__ p, int n) {
    int i = blockIdx.x * blockDim.x + threadIdx.x;
    if (i < n) p[i] = 0;
}

__global__ void k_zero_f32(float* __restrict__ p, int n) {
    int i = blockIdx.x * blockDim.x + threadIdx.x;
    if (i < n) p[i] = 0.f;
}

__global__ void k_copy_i32(const int* __restrict__ a, int* __restrict__ b, int n) {
    int i = blockIdx.x * blockDim.x + threadIdx.x;
    if (i < n) b[i] = a[i];
}

// ============================ CSR construction ============================

__global__ void k_hist(const int* __restrict__ dst, int* __restrict__ cnt, int n) {
    int i = blockIdx.x * blockDim.x + threadIdx.x;
    if (i < n) atomicAdd(&cnt[dst[i]], 1);
}

// single-block exclusive scan over n ints; also writes out[n] = total
__global__ __launch_bounds__(1024)
void k_exscan(const int* __restrict__ in, int* __restrict__ out, int n) {
    __shared__ int sh[1024];
    __shared__ int carry;
    const int t = threadIdx.x;
    if (t == 0) carry = 0;
    __syncthreads();
    for (int base = 0; base < n; base += 1024) {
        int i = base + t;
        int v = (i < n) ? in[i] : 0;
        sh[t] = v;
        __syncthreads();
        for (int off = 1; off < 1024; off <<= 1) {
            int add = (t >= off) ? sh[t - off] : 0;
            __syncthreads();
            sh[t] += add;
            __syncthreads();
        }
        int incl  = sh[t];
        int total = sh[1023];
        if (i < n) out[i] = carry + incl - v;   // exclusive
        __syncthreads();
        if (t == 0) carry += total;
        __syncthreads();
    }
    if (t == 0) out[n] = carry;
}

__global__ void k_fill_csr(const int* __restrict__ src, const int* __restrict__ dst,
                           int* __restrict__ cursor, int* __restrict__ col, int n) {
    int i = blockIdx.x * blockDim.x + threadIdx.x;
    if (i < n) {
        int p = atomicAdd(&cursor[dst[i]], 1);
        col[p] = src[i];
    }
}

// ============================ aggregation (GIN, eps = 0) ============================
// one block per node, one thread per 4 features (float4); gather-sum via CSR.
__global__ void k_aggregate4(const float* __restrict__ x, const int* __restrict__ rp,
                             const int* __restrict__ col, float* __restrict__ agg, int D) {
    const int v = blockIdx.x;
    const int t = threadIdx.x;                 // 0 .. D/4-1
    const int beg = rp[v], end = rp[v + 1];
    const v4f* xr = (const v4f*)x;
    const int Dv = D >> 2;
    v4f acc = xr[(size_t)v * Dv + t];
    for (int e = beg; e < end; ++e) {
        int s = col[e];
        if (e + 1 < end)
            __builtin_prefetch(&xr[(size_t)col[e + 1] * Dv + t], 0, 0);
        acc += xr[(size_t)s * Dv + t];
    }
    ((v4f*)agg)[(size_t)v * Dv + t] = acc;
}

// ============================ fp32 WMMA GEMM ============================
// C[M,N] = A[M,K] @ B[K,N] (+ bias[N]) (+ C if accumulate).  Row-major.
// Block tile 64x64, BK=16, 256 threads = 8 wave32 waves; each wave owns two 16x16
// f32 accumulators driven by V_WMMA_F32_16X16X4_F32.
// Tiles are fetched by GLOBAL_LOAD_ASYNC_TO_LDS_B128 into double-buffered LDS.
// A is stored chunked column-block-major: idx(r,c) = (c>>2)*256 + r*4 + (c&3)
//   -> fragment read {A[r][ks], A[r][ks+1]} is one aligned b64, conflict-free
//      (lanes 0-15 hit banks 4*l16+{0,1}, lanes 16-31 hit 4*l16+{2,3}).
// B is row-major 16x64 -> fragment read is a conflict-free ds_load_2addr_b32.
// Steady-state loop is peeled: it always issues the next stage then waits
// `asynccnt <= 2` (async loads complete in order), so there is no per-stage branch.
#define GEMM_BM 64
#define GEMM_BN 64
#define GEMM_BK 16

__global__ __launch_bounds__(256)
void k_gemm_wmma_f32(const float* __restrict__ A, const float* __restrict__ B,
                     float* __restrict__ C, int M, int N, int K,
                     const float* __restrict__ bias, int accumulate) {
    __shared__ float As[2][GEMM_BM * GEMM_BK];   // 2 x 4KB
    __shared__ float Bs[2][GEMM_BK * GEMM_BN];   // 2 x 4KB

    const int tid  = threadIdx.x;
    const int wave = tid >> 5;       // 0..7
    const int lane = tid & 31;
    const int half = lane >> 4;      // 0 or 1
    const int l16  = lane & 15;
    const int wm   = wave & 3;       // M sub-tile 0..3
    const int wn2  = wave >> 2;      // covers N sub-tiles {2*wn2, 2*wn2+1}

    const int block_m = blockIdx.x * GEMM_BM;
    const int block_n = blockIdx.y * GEMM_BN;

    // per-thread async-copy mapping (one b128 each for A and B per stage)
    const int aRow = tid >> 2;                       // 0..63
    const int aCol = (tid & 3) << 2;                 // 0,4,8,12
    const int gmA  = (block_m + aRow < M) ? (block_m + aRow) : (M - 1); // clamp: padded rows never stored
    const int bRow = tid >> 4;                       // 0..15
    const int bCol = (tid & 15) << 2;                // 0..60

    // running global pointers (advance by BK columns / BK rows per stage)
    const float* ga = A + (size_t)gmA * K + aCol;
    const float* gb = B + (size_t)bRow * N + block_n + bCol;
    const size_t gbStep = (size_t)GEMM_BK * N;

    // per-buffer LDS byte offsets, precomputed
    const unsigned aD = (((unsigned)(tid & 3) << 8) + ((unsigned)aRow << 2)) * 4u;
    const unsigned bD = ((((unsigned)bRow << 6) + (unsigned)bCol)) * 4u;
    const unsigned aLds0 = (unsigned)(uintptr_t)&As[0][0] + aD;
    const unsigned aLds1 = (unsigned)(uintptr_t)&As[1][0] + aD;
    const unsigned bLds0 = (unsigned)(uintptr_t)&Bs[0][0] + bD;
    const unsigned bLds1 = (unsigned)(uintptr_t)&Bs[1][0] + bD;

    v8f acc0 = {0.f, 0.f, 0.f, 0.f, 0.f, 0.f, 0.f, 0.f};
    v8f acc1 = {0.f, 0.f, 0.f, 0.f, 0.f, 0.f, 0.f, 0.f};

    const int am  = wm * 16 + l16;             // row within block tile for A frag
    const int bn0 = (wn2 * 2 + 0) * 16 + l16;  // cols within block tile for B frags
    const int bn1 = (wn2 * 2 + 1) * 16 + l16;

    auto compute = [&](const float* Ac, const float* Bc) {
        #pragma unroll
        for (int kk = 0; kk < GEMM_BK; kk += 4) {
            const int ks = kk + 2 * half;
            const int ai = ((ks >> 2) << 8) + (am << 2) + (ks & 3);
            v2f a, b0, b1;
            a.x  = Ac[ai];
            a.y  = Ac[ai + 1];
            b0.x = Bc[ks * GEMM_BN + bn0];
            b0.y = Bc[(ks + 1) * GEMM_BN + bn0];
            b1.x = Bc[ks * GEMM_BN + bn1];
            b1.y = Bc[(ks + 1) * GEMM_BN + bn1];
            acc0 = __builtin_amdgcn_wmma_f32_16x16x4_f32(
                       false, a, false, b0, (short)0, acc0, false, false);
            acc1 = __builtin_amdgcn_wmma_f32_16x16x4_f32(
                       false, a, false, b1, (short)0, acc1, false, false);
        }
    };

    // stage 0 prologue
    async_lds_b128(aLds0, ga);
    async_lds_b128(bLds0, gb);
    ga += GEMM_BK;
    gb += gbStep;

    const int nstages = K / GEMM_BK;
    int cur = 0;
    for (int s = 0; s < nstages - 1; ++s) {
        // issue stage s+1 into the other buffer, then wait for stage s's 2 loads
        async_lds_b128(cur ? aLds0 : aLds1, ga);
        async_lds_b128(cur ? bLds0 : bLds1, gb);
        ga += GEMM_BK;
        gb += gbStep;
        wait_async_2();
        __syncthreads();
        compute(As[cur], Bs[cur]);
        __syncthreads();   // all waves done reading buf before its DMA overwrite
        cur ^= 1;
    }
    wait_async_0();
    __syncthreads();
    compute(As[cur], Bs[cur]);

    // epilogue: bias + optional accumulate into C
    const int gn0 = block_n + bn0;
    const int gn1 = block_n + bn1;
    const float bi0 = bias ? bias[gn0] : 0.f;
    const float bi1 = bias ? bias[gn1] : 0.f;
    #pragma unroll
    for (int e = 0; e < 8; ++e) {
        int m = block_m + wm * 16 + e + 8 * half;
        if (m < M) {
            size_t i0 = (size_t)m * N + gn0;
            size_t i1 = (size_t)m * N + gn1;
            float v0 = acc0[e] + bi0;
            float v1 = acc1[e] + bi1;
            if (accumulate) { v0 += C[i0]; v1 += C[i1]; }
            C[i0] = v0;
            C[i1] = v1;
        }
    }
}

// ============================ batchnorm (train mode, node axis) ============================

// sums[0:256] = column sums, sums[256:512] = column sums of squares
__global__ __launch_bounds__(256)
void k_bn_stats(const float* __restrict__ z, float* __restrict__ sums, int M) {
    const int t = threadIdx.x;          // column
    float s = 0.f, s2 = 0.f;
    for (int r = blockIdx.x; r < M; r += gridDim.x) {
        float v = z[(size_t)r * GIN_HID + t];
        s  += v;
        s2 += v * v;
    }
    atomicAdd(&sums[t], s);
    atomicAdd(&sums[GIN_HID + t], s2);
}

__global__ __launch_bounds__(256)
void k_bn_apply_relu(const float* __restrict__ z, const float* __restrict__ sums,
                     const float* __restrict__ gamma, const float* __restrict__ beta,
                     float* __restrict__ y, int M) {
    const int t = threadIdx.x;
    const float inv  = 1.0f / (float)M;
    const float mean = sums[t] * inv;
    const float var  = sums[GIN_HID + t] * inv - mean * mean;   // biased variance
    const float sc   = gamma[t] * rsqrtf(var + GIN_BN_EPS);
    const float bi   = beta[t] - mean * sc;
    for (int r = blockIdx.x; r < M; r += gridDim.x) {
        float v = z[(size_t)r * GIN_HID + t] * sc + bi;
        y[(size_t)r * GIN_HID + t] = v > 0.f ? v : 0.f;
    }
}

// ============================ launch ============================

extern "C" void kernel_launch(void* const* d_in, const int* in_sizes, int n_in,
                              void* d_out, int out_size, void* d_ws, size_t ws_size,
                              hipStream_t stream) {
    (void)in_sizes; (void)n_in; (void)out_size; (void)ws_size;

    const float* h        = (const float*)d_in[0];
    const float* conv0_W1 = (const float*)d_in[1];
    const float* convs_W1 = (const float*)d_in[2];  // [3,256,256]
    const float* convs_W2 = (const float*)d_in[3];  // [4,256,256]
    const float* bn1_g    = (const float*)d_in[4];
    const float* bn1_b    = (const float*)d_in[5];
    const float* bn2_g    = (const float*)d_in[6];
    const float* bn2_b    = (const float*)d_in[7];
    const float* pred0_W  = (const float*)d_in[8];
    const float* pred0_b  = (const float*)d_in[9];
    const float* preds_W  = (const float*)d_in[10]; // [4,256,64]
    const float* preds_b  = (const float*)d_in[11]; // [4,64]
    const int*   src      = (const int*)d_in[12];
    const int*   dst      = (const int*)d_in[13];
    float*       out      = (float*)d_out;

    const int Nn = GIN_N_NODES, Ne = GIN_N_EDGES;
    const size_t ACT = (size_t)Nn * GIN_HID;

    // workspace layout (~208 MB)
    float* xa      = (float*)d_ws;
    float* xb      = xa  + ACT;
    float* agg     = xb  + ACT;
    float* z       = agg + ACT;
    float* sums    = z   + ACT;                // 512 floats
    int*   row_ptr = (int*)(sums + 512);       // Nn+1
    int*   cursor  = row_ptr + (Nn + 1);       // Nn (also used as histogram)
    int*   col     = cursor + Nn;              // Ne

    // ---- CSR build ----
    k_zero_i32<<<(Nn + 255) / 256, 256, 0, stream>>>(cursor, Nn);
    k_hist<<<(Ne + 255) / 256, 256, 0, stream>>>(dst, cursor, Ne);
    k_exscan<<<1, 1024, 0, stream>>>(cursor, row_ptr, Nn);
    k_copy_i32<<<(Nn + 255) / 256, 256, 0, stream>>>(row_ptr, cursor, Nn);
    k_fill_csr<<<(Ne + 255) / 256, 256, 0, stream>>>(src, dst, cursor, col, Ne);

    const dim3 gemm_block(256);
    const int  mg = (Nn + GEMM_BM - 1) / GEMM_BM;

    // ---- head 0: out = h @ pred0_W + pred0_b (writes every output element) ----
    k_gemm_wmma_f32<<<dim3(mg, GIN_OUT_DIM / GEMM_BN), gemm_block, 0, stream>>>(
        h, pred0_W, out, Nn, GIN_OUT_DIM, GIN_IN_DIM, pred0_b, 0);

    // ---- conv layers ----
    const float* xp = h;
    int D = GIN_IN_DIM;
    for (int i = 0; i < GIN_N_CONV; ++i) {
        float* xcur = (i & 1) ? xb : xa;

        // agg = x + sum_{src->v} x[src]
        k_aggregate4<<<Nn, D / 4, 0, stream>>>(xp, row_ptr, col, agg, D);

        // z = agg @ W1_i
        const float* W1 = (i == 0) ? conv0_W1 : (convs_W1 + (size_t)(i - 1) * GIN_HID * GIN_HID);
        k_gemm_wmma_f32<<<dim3(mg, GIN_HID / GEMM_BN), gemm_block, 0, stream>>>(
            agg, W1, z, Nn, GIN_HID, D, nullptr, 0);

        // z = relu(bn1(z))   (in place)
        k_zero_f32<<<2, 256, 0, stream>>>(sums, 512);
        k_bn_stats<<<512, 256, 0, stream>>>(z, sums, Nn);
        k_bn_apply_relu<<<512, 256, 0, stream>>>(z, sums, bn1_g + i * GIN_HID,
                                                 bn1_b + i * GIN_HID, z, Nn);

        // agg = z @ W2_i  (reuse agg buffer)
        k_gemm_wmma_f32<<<dim3(mg, GIN_HID / GEMM_BN), gemm_block, 0, stream>>>(
            z, convs_W2 + (size_t)i * GIN_HID * GIN_HID, agg, Nn, GIN_HID, GIN_HID, nullptr, 0);

        // xcur = relu(bn2(agg))
        k_zero_f32<<<2, 256, 0, stream>>>(sums, 512);
        k_bn_stats<<<512, 256, 0, stream>>>(agg, sums, Nn);
        k_bn_apply_relu<<<512, 256, 0, stream>>>(agg, sums, bn2_g + i * GIN_HID,
                                                 bn2_b + i * GIN_HID, xcur, Nn);

        // out += xcur @ preds_W[i] + preds_b[i]
        k_gemm_wmma_f32<<<dim3(mg, GIN_OUT_DIM / GEMM_BN), gemm_block, 0, stream>>>(
            xcur, preds_W + (size_t)i * GIN_HID * GIN_OUT_DIM, out,
            Nn, GIN_OUT_DIM, GIN_HID, preds_b + i * GIN_OUT_DIM, 1);

        xp = xcur;
        D = GIN_HID;
    }
}